// conv_capsules_33225867002119
// MI455X (gfx1250) — compile-verified
//
#include <hip/hip_runtime.h>
#include <hip/hip_bf16.h>
#include <stdint.h>

// conv_capsules patch extraction on MI455X via the Tensor Data Mover (TDM).
//
// out[(b,oh,ow,kh,kw,c,d)] = x[(b, oh+kh, ow+kw, c, d)]
//
// Per (b,oh,ow): the 9KB output slab [kh,kw,c,d] is contiguous, and its source
// is a 2D tile of x: 3 lines of 3KB (kw,c,d contiguous), line stride 64KB (kh).
// TENSOR_LOAD_TO_LDS lands the tile in LDS already in output order;
// TENSOR_STORE_FROM_LDS streams the 9KB out linearly. TDM does all address
// generation; the wave issues 2 tensor ops per 9KB. TENSORcnt orders them.

#define B_   8
#define H_   64
#define W_   64
#define C_   16
#define D_   16
#define KH_  3
#define KW_  3
#define OH_  62
#define OW_  62

#define ROW_ELEMS   (C_ * D_)            // 256 f32 = 1KB, contiguous (c,d) slab
#define LINE_ELEMS  (KW_ * ROW_ELEMS)    // 768 f32 = 3KB  (kw,c,d)
#define TILE_ELEMS  (KH_ * LINE_ELEMS)   // 2304 f32 = 9KB (kh,kw,c,d)
#define STRIDE_ELEMS (W_ * ROW_ELEMS)    // 16384 f32 = 64KB between kh lines

typedef unsigned int v4u __attribute__((ext_vector_type(4)));
typedef int          v8i __attribute__((ext_vector_type(8)));
typedef int          v4i __attribute__((ext_vector_type(4)));

// D# group 0 (ISA §8.3): count=1 (valid), lds_addr[63:32],
// global_addr bits[120:64], type=2 ("image") in bits[127:126].
__device__ static inline v4u tdm_group0(unsigned lds_addr, unsigned long long gaddr) {
    v4u g0;
    g0.x = 1u;                                          // count=1, is_restore=0, gather=0
    g0.y = lds_addr;                                    // LDS byte address
    g0.z = (unsigned)(gaddr & 0xFFFFFFFFull);           // global_addr[31:0]
    g0.w = (unsigned)((gaddr >> 32) & 0x01FFFFFFull)    // global_addr[56:32]
         | 0x80000000u;                                 // type=2
    return g0;
}

// D# group 1 (ISA §8.4): workgroup_mask=0, data_size=2 (4B), no barrier/iterate/pad,
// tensor_dim0/1, tile_dim0/1/2, tensor_dim0_stride (48b), tensor_dim1_stride (48b).
__device__ static inline v8i tdm_group1(unsigned tensor_d0, unsigned tensor_d1,
                                        unsigned tile_d0, unsigned tile_d1,
                                        unsigned long long d0_stride) {
    v8i g1;
    g1[0] = (int)(2u << 16);                                            // data_size=4B
    g1[1] = (int)((tensor_d0 & 0xFFFFu) << 16);                         // dim0[15:0]
    g1[2] = (int)((tensor_d0 >> 16) | ((tensor_d1 & 0xFFFFu) << 16));   // dim0[31:16] | dim1[15:0]
    g1[3] = (int)((tensor_d1 >> 16) | (tile_d0 << 16));                 // dim1[31:16] | tile_dim0
    g1[4] = (int)(tile_d1 & 0xFFFFu);                                   // tile_dim1 | tile_dim2=0
    g1[5] = (int)(d0_stride & 0xFFFFFFFFull);                           // dim0_stride[31:0]
    g1[6] = (int)((d0_stride >> 32) & 0xFFFFull);                       // dim0_stride[47:32] | dim1_stride[15:0]=0
    g1[7] = 0;                                                          // dim1_stride[47:16]=0
    return g1;
}

__global__ __launch_bounds__(32)
void conv_capsules_patch_tdm(const float* __restrict__ x,
                             float* __restrict__ out)
{
    __shared__ __align__(1024) unsigned char smem[TILE_ELEMS * 4];   // 9KB staging

    const int ow = (int)blockIdx.x;      // 0..61
    const int oh = (int)blockIdx.y;      // 0..61
    const int b  = (int)blockIdx.z;      // 0..7

    // Low 32 bits of the generic pointer == addrspace(3) byte offset.
    const unsigned lds0 = (unsigned)(uintptr_t)(&smem[0]);

    // Input tile origin: x[b, oh, ow, 0, 0]; kh lines are +STRIDE_ELEMS apart.
    const unsigned long long xg =
        (unsigned long long)(uintptr_t)x +
        (unsigned long long)(((b * H_ + oh) * W_) + ow) * (ROW_ELEMS * 4ull);

    // Output slab origin: 9 consecutive 1KB chunks at chunk id ((b*OH+oh)*OW+ow)*9.
    const unsigned long long og =
        (unsigned long long)(uintptr_t)out +
        (unsigned long long)(((b * OH_ + oh) * OW_ + ow) * (KH_ * KW_)) * (ROW_ELEMS * 4ull);

    const v4i zero4 = {0, 0, 0, 0};
    const v8i zero8 = {0, 0, 0, 0, 0, 0, 0, 0};

    // Load: 2D tile, 3 lines x 768 elems, line stride 16384 elems.
    {
        const v4u g0 = tdm_group0(lds0, xg);
        const v8i g1 = tdm_group1(/*tensor_d0=*/LINE_ELEMS, /*tensor_d1=*/KH_,
                                  /*tile_d0=*/LINE_ELEMS,   /*tile_d1=*/KH_,
                                  /*d0_stride=*/STRIDE_ELEMS);
        __builtin_amdgcn_tensor_load_to_lds(g0, g1, zero4, zero4, zero8, 0);
    }

    // Tensor ops of one wave are issued in order, but wait for the load's data
    // to be resident in LDS before the store descriptor reads it.
    __builtin_amdgcn_s_wait_tensorcnt(0);

    // Store: 1D run of 2304 elems (9KB), LDS order == output order.
    {
        const v4u g0 = tdm_group0(lds0, og);
        const v8i g1 = tdm_group1(/*tensor_d0=*/TILE_ELEMS, /*tensor_d1=*/1,
                                  /*tile_d0=*/TILE_ELEMS,   /*tile_d1=*/0,
                                  /*d0_stride=*/TILE_ELEMS);
        __builtin_amdgcn_tensor_store_from_lds(g0, g1, zero4, zero4, zero8, 0);
    }
    // S_ENDPGM performs an implicit wait-idle, draining TENSORcnt before the
    // wave (and its LDS allocation) retires.
}

extern "C" void kernel_launch(void* const* d_in, const int* in_sizes, int n_in,
                              void* d_out, int out_size, void* d_ws, size_t ws_size,
                              hipStream_t stream)
{
    (void)in_sizes; (void)n_in; (void)out_size; (void)d_ws; (void)ws_size;

    const float* x = (const float*)d_in[0];
    float* out = (float*)d_out;

    // One 9KB tile per single-wave workgroup; all indexing from blockIdx (SALU).
    // 62*62*8 = 30,752 tiles -> ~283MB streamed, HBM-bound.
    hipLaunchKernelGGL(conv_capsules_patch_tdm,
                       dim3(OW_, OH_, B_), dim3(32), 0, stream,
                       x, out);
}